// FeatureAlign_16260746183044
// MI455X (gfx1250) — compile-verified
//
#include <hip/hip_runtime.h>
#include <math.h>

// ---------------------------------------------------------------------------
// FeatureAlign for MI455X (gfx1250, wave32).
// Dense contractions run on V_WMMA_F32_16X16X4_F32 (fp32 semantics, matrix
// pipe).  Per-block shared operands are staged in LDS (b128 in, ds_load_b64
// fragment reads); strides are compile-time so global loads use immediate
// offsets.  Boundary handling in the 3x3 conv is branchless (clamped address
// + 0/1 multiply) so loads are unconditional and clusterable.  K-loops use
// bounded unrolling (8) to stay inside the VGPR budget -- no scratch spills.
// ---------------------------------------------------------------------------

#define EPSV 1e-5f

typedef __attribute__((ext_vector_type(2))) float v2f;
typedef __attribute__((ext_vector_type(4))) float f4;
typedef __attribute__((ext_vector_type(8))) float v8f;

// D(16x16) += A(16x4) * B(4x16), fp32.
// Lane l (lo=l&15, hi=l>>4):
//   A: a[j] = A[m=lo][k=2*hi+j]
//   B: b[j] = B[k=2*hi+j][n=lo]
//   D: d[j] = D[m=j+8*hi][n=lo]
__device__ __forceinline__ v8f wmma4(v2f a, v2f b, v8f c) {
  return __builtin_amdgcn_wmma_f32_16x16x4_f32(false, a, false, b, (short)0, c,
                                               false, false);
}

__device__ __forceinline__ float silu_f(float y) {
  return y / (1.0f + __expf(-y));
}

// Stage a 16x128 fp32 tile (row-major, 8 KB) into LDS with two b128 loads
// per thread (256 threads).  Barrier included.
__device__ __forceinline__ void stage16x128(const float* __restrict__ g,
                                            float* s) {
  const f4* gv = (const f4*)g;
  f4* sv = (f4*)s;
  int tid = threadIdx.x;
  sv[tid] = gv[tid];
  sv[tid + 256] = gv[tid + 256];
  __syncthreads();
}

// K=128 GEMM micro-kernel: A fragments from LDS (16x128 tile), B fragments
// from a per-lane global weight row (unit stride) -> 32 WMMAs.
__device__ __forceinline__ v8f gemm128_staged(const float* s,
                                              const float* __restrict__ wrow,
                                              int lo, int hi) {
  v8f acc = {0, 0, 0, 0, 0, 0, 0, 0};
  const float* arow = s + lo * 128 + 2 * hi;
  const float* brow = wrow + 2 * hi;
#pragma unroll 8
  for (int kb = 0; kb < 128; kb += 4) {
    v2f a, b;
    a.x = arow[kb];
    a.y = arow[kb + 1];
    b.x = brow[kb];
    b.y = brow[kb + 1];
    acc = wmma4(a, b, acc);
  }
  return acc;
}

// ---------------------------------------------------------------------------
// Kernel 1: 1x1 conv + BN(eval) + SiLU, NCHW -> NCHW.  Co = 128.
// Block = one 16-pixel tile; its CI x 16 input slab is staged in LDS and
// shared by the 8 co-tile waves.  Weights stream from global (unit stride,
// L2-resident).  CI/HW are template constants -> immediate offsets.
// ---------------------------------------------------------------------------
template <int CI, int HW>
__global__ void k_conv1x1_bn_silu(const float* __restrict__ in,
                                  const float* __restrict__ w,
                                  const float* __restrict__ g,
                                  const float* __restrict__ be,
                                  const float* __restrict__ rm,
                                  const float* __restrict__ rv,
                                  float* __restrict__ out) {
  __shared__ float Bs[CI * 16];  // [k][n]
  const int nPxT = HW / 16;
  int tid = threadIdx.x;
  int blk = blockIdx.x;
  int pxt = blk % nPxT;
  int n   = blk / nPxT;
  int px0 = pxt * 16;

  // stage B tile (CI x 16): rows of 16 pixels are 64B-contiguous
  const float* gsrc = in + (size_t)n * CI * HW + px0;
  for (int l = tid; l < CI * 16; l += 256) {
    int k = l >> 4, c = l & 15;
    Bs[l] = gsrc[(size_t)k * HW + c];
  }
  __syncthreads();

  int lane = tid & 31;
  int lo = lane & 15, hi = lane >> 4;
  int co0 = (tid >> 5) * 16;

  v8f acc = {0, 0, 0, 0, 0, 0, 0, 0};
  const float* ap = w + (size_t)(co0 + lo) * CI + 2 * hi;
  const float* bp = Bs + (2 * hi) * 16 + lo;
#pragma unroll 8
  for (int kb = 0; kb < CI; kb += 4) {
    v2f a, b;
    a.x = ap[kb];
    a.y = ap[kb + 1];
    b.x = bp[kb * 16];
    b.y = bp[kb * 16 + 16];
    acc = wmma4(a, b, acc);
  }

  size_t outBase = (size_t)n * 128 * HW + (size_t)px0 + lo;
  for (int j = 0; j < 8; ++j) {
    int co = co0 + j + 8 * hi;
    float s  = g[co] * rsqrtf(rv[co] + EPSV);
    float bb = be[co] - rm[co] * s;
    float y  = acc[j] * s + bb;
    out[outBase + (size_t)co * HW] = silu_f(y);
  }
}

// ---------------------------------------------------------------------------
// Kernel 2: 3x3 stride-2 pad-1 conv + BN + SiLU.
// in: (4,128,192,192) NCHW   out: (4,96,96,128) NHWC
// 9 shifted tap-GEMMs, 32 WMMAs each.  Boundary columns use a clamped
// address + 0/1 scale (branchless; no per-load EXEC masking).
// ---------------------------------------------------------------------------
__global__ void k_conv3x3s2_bn_silu(const float* __restrict__ in,
                                    const float* __restrict__ w,
                                    const float* __restrict__ g,
                                    const float* __restrict__ be,
                                    const float* __restrict__ rm,
                                    const float* __restrict__ rv,
                                    float* __restrict__ out) {
  const int Ci = 128, Hi = 192, Wi = 192;
  int wid  = (int)((blockIdx.x * blockDim.x + threadIdx.x) >> 5);
  int lane = threadIdx.x & 31;
  int lo = lane & 15, hi = lane >> 4;
  int cot = wid & 7;
  int t   = wid >> 3;
  int xt  = t % 6;  t /= 6;
  int oy  = t % 96;
  int n   = t / 96;
  int co0 = cot * 16, ox0 = xt * 16;
  int ixl = 2 * (ox0 + lo) - 1;  // input x for this lane, before +kx

  v8f acc = {0, 0, 0, 0, 0, 0, 0, 0};
  for (int ky = 0; ky < 3; ++ky) {
    int iy = 2 * oy - 1 + ky;
    if (iy < 0 || iy >= Hi) continue;  // uniform over the wave
    for (int kx = 0; kx < 3; ++kx) {
      int ix    = ixl + kx;
      int ixc   = min(max(ix, 0), Wi - 1);              // clamped (in-bounds)
      float okf = (ix >= 0 && ix < Wi) ? 1.0f : 0.0f;   // 0/1 validity
      const float* bp =
          in + (((size_t)n * Ci) * Hi + iy) * (size_t)Wi + ixc +
          (size_t)(2 * hi) * Hi * Wi;
      const float* ap = w + ((size_t)(co0 + lo) * Ci + 2 * hi) * 9 + ky * 3 + kx;
#pragma unroll 8
      for (int kb = 0; kb < Ci; kb += 4) {
        v2f a, b;
        a.x = ap[kb * 9];
        a.y = ap[kb * 9 + 9];
        b.x = bp[(size_t)kb * Hi * Wi] * okf;
        b.y = bp[(size_t)(kb + 1) * Hi * Wi] * okf;
        acc = wmma4(a, b, acc);
      }
    }
  }
  size_t ob = (((size_t)n * 96 + oy) * 96 + ox0 + lo) * 128;
  for (int j = 0; j < 8; ++j) {
    int co = co0 + j + 8 * hi;
    float s  = g[co] * rsqrtf(rv[co] + EPSV);
    float bb = be[co] - rm[co] * s;
    float y  = acc[j] * s + bb;
    out[ob + co] = silu_f(y);
  }
}

// ---------------------------------------------------------------------------
// Kernel 3: 2x bilinear upsample (half-pixel, edge clamp)
// in: (4,128,48,48) NCHW    out: (4,96,96,128) NHWC
// ---------------------------------------------------------------------------
__global__ void k_upsample2x(const float* __restrict__ in,
                             float* __restrict__ out) {
  int tid  = (int)(blockIdx.x * blockDim.x + threadIdx.x);
  int lane = tid & 31;
  int pix  = tid >> 5;
  if (pix >= 4 * 96 * 96) return;
  int ox = pix % 96;
  int tt = pix / 96;
  int oy = tt % 96;
  int n  = tt / 96;
  float sy = (oy + 0.5f) * 0.5f - 0.5f;
  float sx = (ox + 0.5f) * 0.5f - 0.5f;
  float yf = floorf(sy), xf = floorf(sx);
  float wy = sy - yf, wx = sx - xf;
  int y0 = (int)yf, x0 = (int)xf;
  int y0c = min(max(y0, 0), 47), y1c = min(max(y0 + 1, 0), 47);
  int x0c = min(max(x0, 0), 47), x1c = min(max(x0 + 1, 0), 47);
  size_t base = (size_t)n * 128 * 2304;
  size_t ob   = (size_t)pix * 128 + lane * 4;
  for (int j = 0; j < 4; ++j) {
    int c = lane * 4 + j;
    const float* p = in + base + (size_t)c * 2304;
    float v00 = p[y0c * 48 + x0c], v01 = p[y0c * 48 + x1c];
    float v10 = p[y1c * 48 + x0c], v11 = p[y1c * 48 + x1c];
    out[ob + j] = (1.0f - wy) * ((1.0f - wx) * v00 + wx * v01) +
                  wy * ((1.0f - wx) * v10 + wx * v11);
  }
}

// ---------------------------------------------------------------------------
// Kernel 4: DCNv3 input projection: out[Rx128] = in[Rx128] @ W^T + b.
// Block = one 16-row tile staged in LDS, 8 waves over col-tiles.
// ---------------------------------------------------------------------------
__global__ void k_proj128(const float* __restrict__ in,
                          const float* __restrict__ w,
                          const float* __restrict__ bias,
                          float* __restrict__ out) {
  __shared__ float As[16 * 128];
  int r0 = blockIdx.x * 16;
  stage16x128(in + (size_t)r0 * 128, As);

  int lane = threadIdx.x & 31;
  int lo = lane & 15, hi = lane >> 4;
  int co0 = (threadIdx.x >> 5) * 16;

  v8f acc = gemm128_staged(As, w + (size_t)(co0 + lo) * 128, lo, hi);

  float bv = bias[co0 + lo];
  for (int j = 0; j < 8; ++j) {
    int r = r0 + j + 8 * hi;
    out[(size_t)r * 128 + co0 + lo] = acc[j] + bv;
  }
}

// ---------------------------------------------------------------------------
// Kernel 5: depthwise 3x3 conv + bias + LayerNorm + GELU(tanh), NHWC.
// One wave per pixel, 4 channels per lane, wave-shuffle LN reduction.
// ---------------------------------------------------------------------------
__global__ void k_dw_ln_gelu(const float* __restrict__ x,
                             const float* __restrict__ dww,
                             const float* __restrict__ dwb,
                             const float* __restrict__ lng,
                             const float* __restrict__ lnb,
                             float* __restrict__ out) {
  int wid  = (int)((blockIdx.x * blockDim.x + threadIdx.x) >> 5);
  int lane = threadIdx.x & 31;
  int xc = wid % 96;
  int tt = wid / 96;
  int yy = tt % 96;
  int n  = tt / 96;

  float v[4] = {0.0f, 0.0f, 0.0f, 0.0f};
  for (int ky = 0; ky < 3; ++ky) {
    int iy = yy + ky - 1;
    if (iy < 0 || iy >= 96) continue;
    for (int kx = 0; kx < 3; ++kx) {
      int ix = xc + kx - 1;
      if (ix < 0 || ix >= 96) continue;
      const float* p  = x + (((size_t)n * 96 + iy) * 96 + ix) * 128 + lane * 4;
      const float* wp = dww + (ky * 3 + kx) * 128 + lane * 4;
      for (int j = 0; j < 4; ++j) v[j] += p[j] * wp[j];
    }
  }
  for (int j = 0; j < 4; ++j) v[j] += dwb[lane * 4 + j];

  float s  = v[0] + v[1] + v[2] + v[3];
  float s2 = v[0] * v[0] + v[1] * v[1] + v[2] * v[2] + v[3] * v[3];
  for (int m = 16; m > 0; m >>= 1) {
    s  += __shfl_xor(s, m, 32);
    s2 += __shfl_xor(s2, m, 32);
  }
  float mean = s * (1.0f / 128.0f);
  float var  = s2 * (1.0f / 128.0f) - mean * mean;
  float inv  = rsqrtf(var + EPSV);

  float* o = out + (size_t)wid * 128 + lane * 4;
  for (int j = 0; j < 4; ++j) {
    float tn = (v[j] - mean) * inv * lng[lane * 4 + j] + lnb[lane * 4 + j];
    float t3 = tn * tn * tn;
    o[j] = 0.5f * tn *
           (1.0f + tanhf(0.7978845608028654f * (tn + 0.044715f * t3)));
  }
}

// ---------------------------------------------------------------------------
// Kernel 6: fused offset(144) + mask-logit(72) GEMM from dw features.
// 2 blocks per row tile (col tiles 0-7 / 8-13); row tile staged in LDS.
// ---------------------------------------------------------------------------
__global__ void k_offmask(const float* __restrict__ in,
                          const float* __restrict__ offw,
                          const float* __restrict__ offb,
                          const float* __restrict__ mw,
                          const float* __restrict__ mb,
                          float* __restrict__ off,
                          float* __restrict__ mlog) {
  __shared__ float As[16 * 128];
  int rt   = blockIdx.x >> 1;
  int half = blockIdx.x & 1;
  int r0   = rt * 16;
  stage16x128(in + (size_t)r0 * 128, As);

  int nt = half * 8 + (int)(threadIdx.x >> 5);
  if (nt >= 14) return;  // whole-wave exit after barrier
  int lane = threadIdx.x & 31;
  int lo = lane & 15, hi = lane >> 4;
  int cn  = nt * 16 + lo;               // output column 0..223
  int cnc = cn < 216 ? cn : 215;        // clamp pad lanes to a valid row
  const float* wrow = (cnc < 144) ? offw + (size_t)cnc * 128
                                  : mw + (size_t)(cnc - 144) * 128;

  v8f acc = gemm128_staged(As, wrow, lo, hi);

  float bv = (cnc < 144) ? offb[cnc] : mb[cnc - 144];
  for (int j = 0; j < 8; ++j) {
    int r = r0 + j + 8 * hi;
    float val = acc[j] + bv;
    if (cn < 144)
      off[(size_t)r * 144 + cn] = val;
    else if (cn < 216)
      mlog[(size_t)r * 72 + (cn - 144)] = val;
  }
}

// ---------------------------------------------------------------------------
// Kernel 7: softmax over P=9 + deformable bilinear gather, per (pixel, group).
// ---------------------------------------------------------------------------
__global__ void k_sample(const float* __restrict__ xp,
                         const float* __restrict__ off,
                         const float* __restrict__ mlog,
                         float* __restrict__ samp) {
  int tid = (int)(blockIdx.x * blockDim.x + threadIdx.x);
  if (tid >= 4 * 96 * 96 * 8) return;
  int g = tid & 7;
  int r = tid >> 3;
  int xc = r % 96;
  int tt = r / 96;
  int yy = tt % 96;
  int n  = tt / 96;

  float lg[9];
  const float* mp = mlog + (size_t)r * 72 + g * 9;
  float mx = -1e30f;
  for (int p = 0; p < 9; ++p) { lg[p] = mp[p]; mx = fmaxf(mx, lg[p]); }
  float sum = 0.0f;
  for (int p = 0; p < 9; ++p) { lg[p] = __expf(lg[p] - mx); sum += lg[p]; }
  float rs = 1.0f / sum;

  f4 a0 = {0, 0, 0, 0}, a1 = {0, 0, 0, 0}, a2 = {0, 0, 0, 0}, a3 = {0, 0, 0, 0};
  const float* ob = off + (size_t)r * 144 + g * 18;
  for (int p = 0; p < 9; ++p) {
    float dx = ob[p * 2 + 0];
    float dy = ob[p * 2 + 1];
    float py = (float)(yy + 1) + (float)(p / 3 - 1) + dy;
    float px = (float)(xc + 1) + (float)(p % 3 - 1) + dx;
    float yf = floorf(py), xf = floorf(px);
    float wy = py - yf, wx = px - xf;
    int y0 = (int)yf, x0 = (int)xf;
    float m = lg[p] * rs;
    float wgt[4] = {(1.0f - wy) * (1.0f - wx) * m, (1.0f - wy) * wx * m,
                    wy * (1.0f - wx) * m, wy * wx * m};
#pragma unroll
    for (int c = 0; c < 4; ++c) {
      int yi = y0 + (c >> 1);
      int xi = x0 + (c & 1);
      if (yi >= 1 && yi <= 96 && xi >= 1 && xi <= 96) {
        const f4* v = (const f4*)(xp + ((((size_t)n * 96 + (yi - 1)) * 96 +
                                         (xi - 1)) * 128 + g * 16));
        float wv = wgt[c];
        a0 += v[0] * wv; a1 += v[1] * wv; a2 += v[2] * wv; a3 += v[3] * wv;
      }
    }
  }
  f4* o = (f4*)(samp + (size_t)r * 128 + g * 16);
  o[0] = a0; o[1] = a1; o[2] = a2; o[3] = a3;
}

// ---------------------------------------------------------------------------
// Kernel 8: output projection + ReLU -> NCHW slice of (4,384,96,96) concat.
// ---------------------------------------------------------------------------
__global__ void k_outproj_relu(const float* __restrict__ in,
                               const float* __restrict__ w,
                               const float* __restrict__ bias,
                               float* __restrict__ out, int chofs) {
  __shared__ float As[16 * 128];
  int r0 = blockIdx.x * 16;
  stage16x128(in + (size_t)r0 * 128, As);

  int lane = threadIdx.x & 31;
  int lo = lane & 15, hi = lane >> 4;
  int co0 = (threadIdx.x >> 5) * 16;

  v8f acc = gemm128_staged(As, w + (size_t)(co0 + lo) * 128, lo, hi);

  int co   = co0 + lo;
  float bv = bias[co];
  for (int j = 0; j < 8; ++j) {
    int r  = r0 + j + 8 * hi;
    int xc = r % 96;
    int tt = r / 96;
    int yy = tt % 96;
    int n  = tt / 96;
    float v = fmaxf(acc[j] + bv, 0.0f);
    out[(((size_t)n * 384 + chofs + co) * 96 + yy) * 96 + xc] = v;
  }
}

// ---------------------------------------------------------------------------
// Kernel 9: copy x0 into channels 128..255 of the concat output.
// ---------------------------------------------------------------------------
__global__ void k_copy_x0(const float* __restrict__ x0,
                          float* __restrict__ out) {
  size_t i      = (size_t)blockIdx.x * blockDim.x + threadIdx.x;
  size_t stride = (size_t)gridDim.x * blockDim.x;
  size_t total  = (size_t)4 * 128 * 9216;
  for (; i < total; i += stride) {
    size_t px = i % 9216;
    size_t t  = i / 9216;
    size_t c  = t % 128;
    size_t n  = t / 128;
    out[((n * 384 + 128 + c) * 9216) + px] = x0[i];
  }
}

// ---------------------------------------------------------------------------
extern "C" void kernel_launch(void* const* d_in, const int* in_sizes, int n_in,
                              void* d_out, int out_size, void* d_ws,
                              size_t ws_size, hipStream_t stream) {
  const float* x0       = (const float*)d_in[0];
  const float* x1       = (const float*)d_in[1];
  const float* x2       = (const float*)d_in[2];
  const float* conv_w   = (const float*)d_in[3];
  const float* conv_g   = (const float*)d_in[4];
  const float* conv_b   = (const float*)d_in[5];
  const float* conv_rm  = (const float*)d_in[6];
  const float* conv_rv  = (const float*)d_in[7];
  const float* conv1_w  = (const float*)d_in[8];
  const float* conv1_g  = (const float*)d_in[9];
  const float* conv1_b  = (const float*)d_in[10];
  const float* conv1_rm = (const float*)d_in[11];
  const float* conv1_rv = (const float*)d_in[12];
  const float* conv2_w  = (const float*)d_in[13];
  const float* conv2_g  = (const float*)d_in[14];
  const float* conv2_b  = (const float*)d_in[15];
  const float* conv2_rm = (const float*)d_in[16];
  const float* conv2_rv = (const float*)d_in[17];
  const float* dw_w     = (const float*)d_in[18];
  const float* dw_b     = (const float*)d_in[19];
  const float* ln_g     = (const float*)d_in[20];
  const float* ln_b     = (const float*)d_in[21];
  const float* off_w    = (const float*)d_in[22];
  const float* off_b    = (const float*)d_in[23];
  const float* m_w      = (const float*)d_in[24];
  const float* m_b      = (const float*)d_in[25];
  const float* ip_w     = (const float*)d_in[26];
  const float* ip_b     = (const float*)d_in[27];
  const float* op_w     = (const float*)d_in[28];
  const float* op_b     = (const float*)d_in[29];

  float* ws  = (float*)d_ws;
  float* out = (float*)d_out;

  // Workspace layout (floats).  region0 [0, 22118400) holds y2 (18.9M) and
  // later y1 (1.2M), then is reused as DCNv3 scratch.
  const size_t OFF_XP   = 0;
  const size_t OFF_DW   = 4718592;
  const size_t OFF_OFFS = 9437184;
  const size_t OFF_MLOG = 14745600;
  const size_t OFF_SAMP = 17399808;
  const size_t OFF_FU1  = 22118400;
  const size_t OFF_FU2  = 26836992;

  float* y2  = ws;             // (4,128,192,192) NCHW
  float* y1  = ws;             // (4,128,48,48)   NCHW (after y2 consumed)
  float* fu1 = ws + OFF_FU1;   // (4,96,96,128)   NHWC
  float* fu2 = ws + OFF_FU2;   // (4,96,96,128)   NHWC

  // 1) y2 = conv1(x2) 64->128, 192x192: one block per 16-pixel tile
  k_conv1x1_bn_silu<64, 36864><<<9216, 256, 0, stream>>>(
      x2, conv1_w, conv1_g, conv1_b, conv1_rm, conv1_rv, y2);
  // 2) fu2 = conv2(y2) 3x3 s2 -> 96x96 NHWC: 18432 waves
  k_conv3x3s2_bn_silu<<<2304, 256, 0, stream>>>(y2, conv2_w, conv2_g, conv2_b,
                                                conv2_rm, conv2_rv, fu2);
  // 3) y1 = conv(x1) 256->128, 48x48
  k_conv1x1_bn_silu<256, 2304><<<576, 256, 0, stream>>>(
      x1, conv_w, conv_g, conv_b, conv_rm, conv_rv, y1);
  // 4) fu1 = bilinear 2x upsample of y1 -> NHWC
  k_upsample2x<<<4608, 256, 0, stream>>>(y1, fu1);

  // DCNv3 pipeline (5 kernels), run twice.
  auto dcn = [&](const float* src, int chofs) {
    k_proj128<<<2304, 256, 0, stream>>>(src, ip_w, ip_b, ws + OFF_XP);
    k_dw_ln_gelu<<<4608, 256, 0, stream>>>(src, dw_w, dw_b, ln_g, ln_b,
                                           ws + OFF_DW);
    k_offmask<<<4608, 256, 0, stream>>>(ws + OFF_DW, off_w, off_b, m_w, m_b,
                                        ws + OFF_OFFS, ws + OFF_MLOG);
    k_sample<<<1152, 256, 0, stream>>>(ws + OFF_XP, ws + OFF_OFFS,
                                       ws + OFF_MLOG, ws + OFF_SAMP);
    k_outproj_relu<<<2304, 256, 0, stream>>>(ws + OFF_SAMP, op_w, op_b, out,
                                             chofs);
  };

  dcn(fu1, 0);                                  // channels   0..127
  k_copy_x0<<<4608, 256, 0, stream>>>(x0, out); // channels 128..255
  dcn(fu2, 256);                                // channels 256..383
}